// NCC_weight_46308337385678
// MI455X (gfx1250) — compile-verified
//
#include <hip/hip_runtime.h>
#include <stdint.h>

// Problem geometry (SHAPE = (2, 1, 160, 192, 224), WIN = 5)
#define DZ      160
#define HY      192
#define WX      224
#define NBATCH  2
#define CHUNK   20                        // output z-planes per block
#define NCHUNK  (DZ / CHUNK)              // 8
#define BXO     112                       // output columns per block
#define NQ      (BXO / 4)                 // 28 x-quads per row
#define BY      8                         // output rows per block
#define ROWS    12                        // loaded rows (BY + 4)
#define TLX     128                       // staged row pitch (116 used, 2^7 for exact wave split)
#define NPLANES (CHUNK + 4)               // 24 staged planes per chunk
#define SD      (WX * HY)                 // 43008
#define SB      (SD * DZ)                 // 6881280
#define GX      (WX / BXO)                // 2
#define GY      (HY / BY)                 // 24
#define GZ      (NBATCH * NCHUNK)         // 16
#define NPART   (GX * GY * GZ)            // 768 block partials
#define NTOTAL  ((float)NBATCH * (float)DZ * (float)HY * (float)WX)
#define XTASKS  (ROWS * NQ)               // 336 x-filter tasks per plane

typedef float v4f __attribute__((ext_vector_type(4)));

// ---- CDNA5 async global->LDS path (gfx1250) -------------------------------
// Per-lane HBM -> LDS copy tracked by ASYNCcnt; LDS destination address is the
// low 32 bits of the generic __shared__ pointer (flat->LDS truncation).
__device__ __forceinline__ void async_ld_b32(uint32_t lds_off, const float* g) {
  asm volatile("global_load_async_to_lds_b32 %0, %1, off"
               :: "v"(lds_off), "v"((unsigned long long)(size_t)g)
               : "memory");
}
// Each staged plane issues exactly 12 async loads per wave (uniform count).
__device__ __forceinline__ void wait_async_le12() {
  asm volatile("s_wait_asynccnt 0xC" ::: "memory");
}
__device__ __forceinline__ void wait_async_le0() {
  asm volatile("s_wait_asynccnt 0x0" ::: "memory");
}

__global__ __launch_bounds__(256)
void ncc_fused(const float* __restrict__ I, const float* __restrict__ J,
               const float* __restrict__ Wm, float* __restrict__ partials) {
  __shared__ float ldsI[2][ROWS][TLX];    // double-buffered raw I planes (12 KB)
  __shared__ float ldsJ[2][ROWS][TLX];    // double-buffered raw J planes (12 KB)
  __shared__ float xs[5][ROWS][BXO];      // x-filtered {I,J,I2,J2,IJ}  (26.25 KB)
  __shared__ float red[256];

  const int tx  = threadIdx.x;            // 0..31
  const int ty  = threadIdx.y;            // 0..7
  const int tid = ty * 32 + tx;

  const int b  = blockIdx.z / NCHUNK;
  const int z0 = (blockIdx.z % NCHUNK) * CHUNK;
  const int x0 = blockIdx.x * BXO;
  const int y0 = blockIdx.y * BY;

  // ---- staging map: 12 rows x 128 cols = 1536 cells = 6 slots x 256 threads
  int stage_off[6];
#pragma unroll
  for (int s = 0; s < 6; ++s) {
    const int idx = tid + s * 256;
    const int row = idx >> 7;             // /TLX
    const int col = idx & (TLX - 1);
    const int xg  = x0 - 2 + col;         // clamped for address safety; OOB
    const int yg  = y0 - 2 + row;         // cells are zeroed at consume time
    const int xc  = min(max(xg, 0), WX - 1);
    const int yc  = min(max(yg, 0), HY - 1);
    stage_off[s]  = b * SB + yc * WX + xc;
  }
  const uint32_t ldsI_base = (uint32_t)(size_t)&ldsI[0][0][0];
  const uint32_t ldsJ_base = (uint32_t)(size_t)&ldsJ[0][0][0];

  auto stage = [&](int zg, int buf) {
    const int      zoff = zg * SD;
    const uint32_t bofs = (uint32_t)buf * (ROWS * TLX * 4);
#pragma unroll
    for (int s = 0; s < 6; ++s) {
      const uint32_t lo = bofs + (uint32_t)(tid + s * 256) * 4u;
      async_ld_b32(ldsI_base + lo, I + (zoff + stage_off[s]));
      async_ld_b32(ldsJ_base + lo, J + (zoff + stage_off[s]));
    }
  };

  // ---- x-pass task map: 336 (row, quad) tasks over 256 threads (2 slots)
  int  xp_row[2], xp_quad[2];
  bool xp_on[2];
#pragma unroll
  for (int s = 0; s < 2; ++s) {
    const int task = tid + s * 256;
    xp_on[s] = task < XTASKS;
    const int t = min(task, XTASKS - 1);
    xp_row[s]  = t / NQ;
    xp_quad[s] = t - xp_row[s] * NQ;
  }

  const bool yp_on = (tx < NQ);           // y/z/cc: quad = tx, output row = ty

  v4f hist[5][5];                          // z shift-history per quantity
#pragma unroll
  for (int q = 0; q < 5; ++q)
#pragma unroll
    for (int k = 0; k < 5; ++k) hist[q][k] = (v4f)0.f;
  v4f acc4 = (v4f)0.f;

  if (z0 - 2 >= 0) stage(z0 - 2, 0);      // prologue

  for (int i = 0; i < NPLANES; ++i) {
    const int  zg = z0 - 2 + i;
    const bool plane_ok = (zg >= 0) && (zg < DZ);
    const int  cur = i & 1;

    // Prefetch next plane into the other buffer (block-uniform control).
    bool staged_next = false;
    if (i + 1 < NPLANES) {
      const int zn = zg + 1;
      if (zn >= 0 && zn < DZ) { stage(zn, (i + 1) & 1); staged_next = true; }
    }
    if (staged_next) wait_async_le12(); else wait_async_le0();
    // Barrier (a): every wave finished waiting its own async loads -> whole
    // 'cur' tile valid; also orders last plane's xs reads vs. xs writes below.
    __syncthreads();

    // ---- x-pass: 8 raw values -> 4 sliding 5-tap sums per quantity
#pragma unroll
    for (int s = 0; s < 2; ++s) {
      if (!xp_on[s]) continue;
      const int row = xp_row[s];
      const int qd  = xp_quad[s];
      const int yg  = y0 - 2 + row;
      const bool rowok = plane_ok && (yg >= 0) && (yg < HY);

      float iv[8], jv[8];
      if (rowok) {
        const v4f a0 = *(const v4f*)&ldsI[cur][row][4 * qd];
        const v4f a1 = *(const v4f*)&ldsI[cur][row][4 * qd + 4];
        const v4f b0 = *(const v4f*)&ldsJ[cur][row][4 * qd];
        const v4f b1 = *(const v4f*)&ldsJ[cur][row][4 * qd + 4];
        iv[0]=a0.x; iv[1]=a0.y; iv[2]=a0.z; iv[3]=a0.w;
        iv[4]=a1.x; iv[5]=a1.y; iv[6]=a1.z; iv[7]=a1.w;
        jv[0]=b0.x; jv[1]=b0.y; jv[2]=b0.z; jv[3]=b0.w;
        jv[4]=b1.x; jv[5]=b1.y; jv[6]=b1.z; jv[7]=b1.w;
        const int xgb = x0 - 2 + 4 * qd;   // global x of element 0
        if (xgb < 0 || xgb + 7 >= WX) {    // only edge quads diverge here
#pragma unroll
          for (int k = 0; k < 8; ++k) {
            const bool ok = (unsigned)(xgb + k) < (unsigned)WX;
            iv[k] = ok ? iv[k] : 0.f;
            jv[k] = ok ? jv[k] : 0.f;
          }
        }
      } else {
#pragma unroll
        for (int k = 0; k < 8; ++k) { iv[k] = 0.f; jv[k] = 0.f; }
      }

      float qq[5][8];
#pragma unroll
      for (int k = 0; k < 8; ++k) {
        qq[0][k] = iv[k];         qq[1][k] = jv[k];
        qq[2][k] = iv[k] * iv[k]; qq[3][k] = jv[k] * jv[k];
        qq[4][k] = iv[k] * jv[k];
      }
#pragma unroll
      for (int q = 0; q < 5; ++q) {
        const float t0 = qq[q][0] + qq[q][1] + qq[q][2] + qq[q][3] + qq[q][4];
        const float t1 = t0 + qq[q][5] - qq[q][0];
        const float t2 = t1 + qq[q][6] - qq[q][1];
        const float t3 = t2 + qq[q][7] - qq[q][2];
        v4f o; o.x = t0; o.y = t1; o.z = t2; o.w = t3;
        *(v4f*)&xs[q][row][4 * qd] = o;    // ds_store_b128, 16B aligned
      }
    }
    // Barrier (b): xs visible; also orders raw reads vs. next stage's writes.
    __syncthreads();

    // ---- y-pass (b128 reads), z shift-history, cc, weighted accumulate
    if (yp_on) {
      v4f ys[5] = {(v4f)0.f, (v4f)0.f, (v4f)0.f, (v4f)0.f, (v4f)0.f};
#pragma unroll
      for (int d = 0; d < 5; ++d) {
        const int rr = ty + d;             // window rows for output y = y0+ty
#pragma unroll
        for (int q = 0; q < 5; ++q) ys[q] += *(const v4f*)&xs[q][rr][4 * tx];
      }
#pragma unroll
      for (int q = 0; q < 5; ++q) {
        hist[q][4] = hist[q][3]; hist[q][3] = hist[q][2];
        hist[q][2] = hist[q][1]; hist[q][1] = hist[q][0];
        hist[q][0] = ys[q];
      }
      if (i >= 4) {                        // output plane zo = zg - 2
        v4f S[5];
#pragma unroll
        for (int q = 0; q < 5; ++q)
          S[q] = hist[q][0] + hist[q][1] + hist[q][2] + hist[q][3] + hist[q][4];
        const float inv = 1.0f / 125.0f;
        const v4f uI = S[0] * inv, uJ = S[1] * inv;
        const v4f cross = S[4] - uJ * S[0] - uI * S[1] + uI * uJ * 125.0f;
        const v4f Ivar  = S[2] - 2.0f * uI * S[0] + uI * uI * 125.0f;
        const v4f Jvar  = S[3] - 2.0f * uJ * S[1] + uJ * uJ * 125.0f;
        const v4f cc = (cross * cross) / (Ivar * Jvar + 1e-8f);
        const int zo = zg - 2;
        const int wi = b * SB + zo * SD + (y0 + ty) * WX + (x0 + 4 * tx);
        const v4f wv = __builtin_nontemporal_load((const v4f*)(Wm + wi));
        acc4 += cc * wv;
      }
    }
  }

  // ---- deterministic block reduction
  red[tid] = acc4.x + acc4.y + acc4.z + acc4.w;
  __syncthreads();
  for (int k = 128; k > 0; k >>= 1) {
    if (tid < k) red[tid] += red[tid + k];
    __syncthreads();
  }
  if (tid == 0) {
    const int pid = ((int)blockIdx.z * GY + (int)blockIdx.y) * GX + (int)blockIdx.x;
    partials[pid] = red[0];
  }
}

// Fixed-order final reduction: deterministic (no float atomics).
__global__ __launch_bounds__(256)
void ncc_finalize(const float* __restrict__ partials, float* __restrict__ out) {
  __shared__ float red[256];
  float s = 0.f;
  for (int i = threadIdx.x; i < NPART; i += 256) s += partials[i];
  red[threadIdx.x] = s;
  __syncthreads();
  for (int k = 128; k > 0; k >>= 1) {
    if ((int)threadIdx.x < k) red[threadIdx.x] += red[threadIdx.x + k];
    __syncthreads();
  }
  if (threadIdx.x == 0) out[0] = -red[0] / NTOTAL;
}

extern "C" void kernel_launch(void* const* d_in, const int* in_sizes, int n_in,
                              void* d_out, int out_size, void* d_ws, size_t ws_size,
                              hipStream_t stream) {
  (void)in_sizes; (void)n_in; (void)out_size; (void)ws_size;
  const float* I  = (const float*)d_in[0];
  const float* J  = (const float*)d_in[1];
  const float* Wm = (const float*)d_in[2];
  float* partials = (float*)d_ws;          // NPART floats = 3 KB scratch

  dim3 blk(32, 8, 1);                      // 8 waves (wave32)
  dim3 grd(GX, GY, GZ);                    // 2 x 24 x 16 = 768 blocks
  ncc_fused<<<grd, blk, 0, stream>>>(I, J, Wm, partials);
  ncc_finalize<<<1, 256, 0, stream>>>(partials, (float*)d_out);
}